// PubmedGATV2Conv_87093346828806
// MI455X (gfx1250) — compile-verified
//
#include <hip/hip_runtime.h>
#include <hip/hip_bf16.h>
#include <math.h>

#define IN_CH 500
#define HEADS 8

typedef float v2f __attribute__((ext_vector_type(2)));
typedef float v8f __attribute__((ext_vector_type(8)));

// Order-preserving int key for float atomicMax (monotone map f32 -> i32)
__device__ __forceinline__ int fkey(float f) {
  int b = __float_as_int(f);
  return b >= 0 ? b : (b ^ 0x7fffffff);
}
__device__ __forceinline__ float funkey(int k) {
  return __int_as_float(k >= 0 ? k : (k ^ 0x7fffffff));
}

// ---------------- init buffers for a layer ----------------
__global__ void k_init1(float* agg1, int* nmaxk, float* nsum, int N) {
  int t = blockIdx.x * blockDim.x + threadIdx.x;
  if (t < N * 64) agg1[t] = 0.f;
  if (t < N * 8) { nmaxk[t] = (int)0x80000000; nsum[t] = 0.f; }
}

__global__ void k_init2(float* agg2, int* nmaxk, float* nsum, int N) {
  int t = blockIdx.x * blockDim.x + threadIdx.x;
  if (t < N * 24) agg2[t] = 0.f;
  if (t < N * 8) { nmaxk[t] = (int)0x80000000; nsum[t] = 0.f; }
}

// ---------------- Layer-1 dual GEMM via fp32 WMMA ----------------
// Computes xl = x @ Wl and xr = x @ Wr  (N x 500) @ (500 x 64).
// One block = 16 rows of x staged in LDS (read x from HBM exactly once).
// 8 waves: waves 0-3 -> Wl col tiles 0..3, waves 4-7 -> Wr col tiles 0..3.
__global__ void __launch_bounds__(256) k_gemm1_wmma(
    const float* __restrict__ x, const float* __restrict__ Wl,
    const float* __restrict__ Wr, float* __restrict__ xl,
    float* __restrict__ xr, int N) {
  __shared__ float As[16 * IN_CH];  // 32 KB
  const int row0 = blockIdx.x * 16;

  for (int t = threadIdx.x; t < 16 * IN_CH; t += 256) {
    int r = t / IN_CH;
    int c = t - r * IN_CH;
    int gr = row0 + r;
    As[t] = (gr < N) ? x[(size_t)gr * IN_CH + c] : 0.f;
  }
  __syncthreads();

  const int wave = threadIdx.x >> 5;
  const int lane = threadIdx.x & 31;
  const float* W = (wave & 4) ? Wr : Wl;
  float* out = (wave & 4) ? xr : xl;
  const int col0 = (wave & 3) * 16;
  const int m = lane & 15;            // A row / B,D column index
  const int kb = (lane >> 4) * 2;     // K sub-offset carried by upper half-wave

  v8f acc = {};
  const float* arow = &As[m * IN_CH + kb];
  for (int k = 0; k < IN_CH; k += 4) {
    v2f a;
    a.x = arow[k];
    a.y = arow[k + 1];
    v2f b;
    b.x = W[(size_t)(k + kb) * 64 + col0 + m];
    b.y = W[(size_t)(k + kb + 1) * 64 + col0 + m];
    acc = __builtin_amdgcn_wmma_f32_16x16x4_f32(false, a, false, b, (short)0,
                                                acc, false, false);
  }

  const int rhalf = (lane >> 4) * 8;
  if (row0 + 16 <= N) {
    // whole tile in range (always true when N % 16 == 0): single base,
    // immediate-offset stores, no per-element exec-mask juggling
    float* po = out + (size_t)(row0 + rhalf) * 64 + col0 + m;
#pragma unroll
    for (int i = 0; i < 8; ++i) po[(size_t)i * 64] = acc[i];
  } else {
#pragma unroll
    for (int i = 0; i < 8; ++i) {
      int row = row0 + i + rhalf;
      if (row < N) out[(size_t)row * 64 + col0 + m] = acc[i];
    }
  }
}

// ---------------- Edge pass A: logits + segment max ----------------
// thread = (edge, head). C = channels per head (8 for layer 1, 3 for layer 2).
template <int C>
__global__ void k_edge_logits(const int* __restrict__ ei, int E, int N,
                              const float* __restrict__ xl,
                              const float* __restrict__ xr,
                              const float* __restrict__ att,
                              float* __restrict__ logit,
                              int* __restrict__ nmaxk) {
  int t = blockIdx.x * blockDim.x + threadIdx.x;
  if (t >= (E + N) * HEADS) return;
  int e = t >> 3, h = t & 7;
  int s, d;
  if (e < E) { s = ei[e]; d = ei[E + e]; } else { s = d = e - E; }
  const float* pl = xl + (size_t)s * (HEADS * C) + h * C;
  const float* pr = xr + (size_t)d * (HEADS * C) + h * C;
  const float* pa = att + h * C;
  float acc = 0.f;
  if constexpr (C == 8) {
    // rows are 32B-aligned: use b128 loads (2 per side)
    float4 l0 = *(const float4*)(pl);
    float4 l1 = *(const float4*)(pl + 4);
    float4 r0 = *(const float4*)(pr);
    float4 r1 = *(const float4*)(pr + 4);
    float4 a0 = *(const float4*)(pa);
    float4 a1 = *(const float4*)(pa + 4);
    float v;
    v = l0.x + r0.x; acc = fmaf(a0.x, v > 0.f ? v : 0.2f * v, acc);
    v = l0.y + r0.y; acc = fmaf(a0.y, v > 0.f ? v : 0.2f * v, acc);
    v = l0.z + r0.z; acc = fmaf(a0.z, v > 0.f ? v : 0.2f * v, acc);
    v = l0.w + r0.w; acc = fmaf(a0.w, v > 0.f ? v : 0.2f * v, acc);
    v = l1.x + r1.x; acc = fmaf(a1.x, v > 0.f ? v : 0.2f * v, acc);
    v = l1.y + r1.y; acc = fmaf(a1.y, v > 0.f ? v : 0.2f * v, acc);
    v = l1.z + r1.z; acc = fmaf(a1.z, v > 0.f ? v : 0.2f * v, acc);
    v = l1.w + r1.w; acc = fmaf(a1.w, v > 0.f ? v : 0.2f * v, acc);
  } else {
#pragma unroll
    for (int c = 0; c < C; ++c) {
      float v = pl[c] + pr[c];
      v = v > 0.f ? v : 0.2f * v;  // LeakyReLU(0.2)
      acc = fmaf(pa[c], v, acc);
    }
  }
  logit[t] = acc;
  atomicMax(&nmaxk[d * HEADS + h], fkey(acc));
}

// ---------------- Edge pass B: exp + segment sum (in-place on logit) -----
__global__ void k_edge_exp(const int* __restrict__ ei, int E, int N,
                           float* __restrict__ logit,
                           const int* __restrict__ nmaxk,
                           float* __restrict__ nsum) {
  int t = blockIdx.x * blockDim.x + threadIdx.x;
  if (t >= (E + N) * HEADS) return;
  int e = t >> 3, h = t & 7;
  int d = (e < E) ? ei[E + e] : (e - E);
  float m = funkey(nmaxk[d * HEADS + h]);
  float ex = expf(logit[t] - m);
  logit[t] = ex;  // reuse buffer as exp(logit - max)
  atomicAdd(&nsum[d * HEADS + h], ex);
}

// ---------------- Edge pass C: alpha-weighted scatter-add ----------------
template <int C>
__global__ void k_edge_agg(const int* __restrict__ ei, int E, int N,
                           const float* __restrict__ xl,
                           const float* __restrict__ ex,
                           const float* __restrict__ nsum,
                           float* __restrict__ agg) {
  int t = blockIdx.x * blockDim.x + threadIdx.x;
  if (t >= (E + N) * HEADS) return;
  int e = t >> 3, h = t & 7;
  int s, d;
  if (e < E) { s = ei[e]; d = ei[E + e]; } else { s = d = e - E; }
  float alpha = ex[t] / nsum[d * HEADS + h];
  const float* pl = xl + (size_t)s * (HEADS * C) + h * C;
  float* po = agg + (size_t)d * (HEADS * C) + h * C;
  if constexpr (C == 8) {
    float4 l0 = *(const float4*)(pl);
    float4 l1 = *(const float4*)(pl + 4);
    atomicAdd(&po[0], alpha * l0.x);
    atomicAdd(&po[1], alpha * l0.y);
    atomicAdd(&po[2], alpha * l0.z);
    atomicAdd(&po[3], alpha * l0.w);
    atomicAdd(&po[4], alpha * l1.x);
    atomicAdd(&po[5], alpha * l1.y);
    atomicAdd(&po[6], alpha * l1.z);
    atomicAdd(&po[7], alpha * l1.w);
  } else {
#pragma unroll
    for (int c = 0; c < C; ++c) atomicAdd(&po[c], alpha * pl[c]);
  }
}

// ---------------- bias + ELU (in-place) ----------------
__global__ void k_bias_elu(float* __restrict__ h, const float* __restrict__ b1,
                           int N) {
  int t = blockIdx.x * blockDim.x + threadIdx.x;
  if (t >= N * 64) return;
  float v = h[t] + b1[t & 63];
  h[t] = v > 0.f ? v : expm1f(v);
}

// ---------------- Layer-2 dual GEMM (tiny: N x 64 @ 64 x 24) -------------
__global__ void k_gemm2(const float* __restrict__ h, const float* __restrict__ Wl,
                        const float* __restrict__ Wr, float* __restrict__ xl2,
                        float* __restrict__ xr2, int N) {
  int t = blockIdx.x * blockDim.x + threadIdx.x;
  if (t >= N * 24) return;
  int n = t / 24, j = t - n * 24;
  const float* hr = h + (size_t)n * 64;
  float al = 0.f, ar = 0.f;
#pragma unroll 8
  for (int k = 0; k < 64; ++k) {
    float hv = hr[k];
    al = fmaf(hv, Wl[k * 24 + j], al);
    ar = fmaf(hv, Wr[k * 24 + j], ar);
  }
  xl2[t] = al;
  xr2[t] = ar;
}

// ---------------- mean over heads + bias + softmax ----------------
__global__ void k_final(const float* __restrict__ agg2,
                        const float* __restrict__ b2, float* __restrict__ out,
                        int N) {
  int n = blockIdx.x * blockDim.x + threadIdx.x;
  if (n >= N) return;
  const float* p = agg2 + (size_t)n * 24;
  float z[3];
#pragma unroll
  for (int k = 0; k < 3; ++k) {
    float s = 0.f;
#pragma unroll
    for (int hh = 0; hh < 8; ++hh) s += p[hh * 3 + k];
    z[k] = 0.125f * s + b2[k];
  }
  float m = fmaxf(z[0], fmaxf(z[1], z[2]));
  float e0 = expf(z[0] - m), e1 = expf(z[1] - m), e2 = expf(z[2] - m);
  float inv = 1.f / (e0 + e1 + e2);
  out[(size_t)n * 3 + 0] = e0 * inv;
  out[(size_t)n * 3 + 1] = e1 * inv;
  out[(size_t)n * 3 + 2] = e2 * inv;
}

extern "C" void kernel_launch(void* const* d_in, const int* in_sizes, int n_in,
                              void* d_out, int out_size, void* d_ws,
                              size_t ws_size, hipStream_t stream) {
  const float* x = (const float*)d_in[0];
  const int* ei = (const int*)d_in[1];  // int32 (JAX x64 disabled)
  const float* W1l = (const float*)d_in[2];
  const float* W1r = (const float*)d_in[3];
  const float* att1 = (const float*)d_in[4];
  const float* b1 = (const float*)d_in[5];
  const float* W2l = (const float*)d_in[6];
  const float* W2r = (const float*)d_in[7];
  const float* att2 = (const float*)d_in[8];
  const float* b2 = (const float*)d_in[9];
  float* out = (float*)d_out;

  const int N = in_sizes[0] / IN_CH;
  const int E = in_sizes[1] / 2;
  const int Et = E + N;

  // workspace layout (floats)
  float* w = (float*)d_ws;
  float* xl1 = w;       w += (size_t)N * 64;
  float* xr1 = w;       w += (size_t)N * 64;
  float* agg1 = w;      w += (size_t)N * 64;  // becomes h after bias+ELU
  float* logit = w;     w += (size_t)Et * 8;  // raw logits, then exp values
  int* nmaxk = (int*)w; w += (size_t)N * 8;
  float* nsum = w;      w += (size_t)N * 8;
  float* xl2 = w;       w += (size_t)N * 24;
  float* xr2 = w;       w += (size_t)N * 24;
  float* agg2 = w;      w += (size_t)N * 24;

  const int B = 256;
  const int egrid = (Et * HEADS + B - 1) / B;

  // ---- layer 1 ----
  k_init1<<<(N * 64 + B - 1) / B, B, 0, stream>>>(agg1, nmaxk, nsum, N);
  k_gemm1_wmma<<<(N + 15) / 16, 256, 0, stream>>>(x, W1l, W1r, xl1, xr1, N);
  k_edge_logits<8><<<egrid, B, 0, stream>>>(ei, E, N, xl1, xr1, att1, logit, nmaxk);
  k_edge_exp<<<egrid, B, 0, stream>>>(ei, E, N, logit, nmaxk, nsum);
  k_edge_agg<8><<<egrid, B, 0, stream>>>(ei, E, N, xl1, logit, nsum, agg1);
  k_bias_elu<<<(N * 64 + B - 1) / B, B, 0, stream>>>(agg1, b1, N);

  // ---- layer 2 ----
  k_init2<<<(N * 24 + B - 1) / B, B, 0, stream>>>(agg2, nmaxk, nsum, N);
  k_gemm2<<<(N * 24 + B - 1) / B, B, 0, stream>>>(agg1, W2l, W2r, xl2, xr2, N);
  k_edge_logits<3><<<egrid, B, 0, stream>>>(ei, E, N, xl2, xr2, att2, logit, nmaxk);
  k_edge_exp<<<egrid, B, 0, stream>>>(ei, E, N, logit, nmaxk, nsum);
  k_edge_agg<3><<<egrid, B, 0, stream>>>(ei, E, N, xl2, logit, nsum, agg2);
  k_final<<<(N + B - 1) / B, B, 0, stream>>>(agg2, b2, out, N);
}